// atom_Model_25786983646092
// MI455X (gfx1250) — compile-verified
//
#include <hip/hip_runtime.h>
#include <hip/hip_bf16.h>

#define N_ATOMS  100000
#define N_RESID  12500
#define N_EDGES  800000
#define D        115
#define D_IN     37
#define N_LAYERS 4
#define DPAD     128
#define M_TILES  (N_ATOMS / 16)   // 6250, N_ATOMS divisible by 16

typedef float v2f __attribute__((ext_vector_type(2)));
typedef float v8f __attribute__((ext_vector_type(8)));

// ---------------------------------------------------------------- utilities
__global__ void zero_f4(float4* p, long n4) {
    long i = (long)blockIdx.x * blockDim.x + threadIdx.x;
    if (i < n4) p[i] = make_float4(0.f, 0.f, 0.f, 0.f);
}

// ------------------------------------------------- weight panel preparation
// Builds, for each (layer, phase), a 128x128 zero-padded B panel B[k][n] =
// W[n][k] (since out = x @ W^T), stored pre-swizzled so that the WMMA B
// fragment (lane reads B[k][n], B[k+1][n]) is one contiguous float2:
//   index(k,n) = (k>>1)*256 + n*2 + (k&1)
__global__ void prep_weights(const float* __restrict__ Wl,
                             const float* __restrict__ Wr,
                             float* __restrict__ wpad) {
    int idx = blockIdx.x * 256 + threadIdx.x;     // < 4*2*128*128
    int n = idx & 127;
    int k = (idx >> 7) & 127;
    int p = (idx >> 14) & 1;
    int l = idx >> 15;
    float v = 0.f;
    if (k < D && n < D) {
        const float* W = p ? Wr : Wl;
        v = W[((long)l * D + n) * D + k];
    }
    wpad[((long)(l * 2 + p) << 14) + ((k >> 1) << 8) + (n << 1) + (k & 1)] = v;
}

// ------------------------------------------------------- input projection
// x[atom][col] = onehot[atom] . ln_W[col] + ln_b[col]  (cols 115..127 -> 0)
__global__ void input_proj(const float* __restrict__ onehot,
                           const float* __restrict__ lnW,
                           const float* __restrict__ lnb,
                           float* __restrict__ x) {
    long t = (long)blockIdx.x * 256 + threadIdx.x;  // exactly N_ATOMS*DPAD
    int col = (int)(t & 127);
    long atom = t >> 7;
    float acc = 0.f;
    if (col < D) {
        acc = lnb[col];
        const float* w = lnW + (long)col * D_IN;
        const float* a = onehot + atom * D_IN;
#pragma unroll
        for (int j = 0; j < D_IN; ++j) acc = fmaf(a[j], w[j], acc);
    }
    x[atom * DPAD + col] = acc;
}

// ----------------------------------------------------------------- degrees
__global__ void deg_count(const int* __restrict__ ei, float* deg) {
    int e = blockIdx.x * 256 + threadIdx.x;   // exactly N_EDGES
    atomicAdd(&deg[ei[N_EDGES + e]], 1.0f);
}

__global__ void deg_inv(float* deg) {
    int i = blockIdx.x * 256 + threadIdx.x;
    if (i < N_ATOMS) deg[i] = 1.0f / fmaxf(deg[i], 1.0f);
}

// --------------------------------------------------------- edge scatter-add
// One wave per edge: float4 gather of x[src], 4 atomic adds into agg[dst].
__global__ void edge_scatter(const int* __restrict__ ei,
                             const float* __restrict__ x,
                             float* __restrict__ agg) {
    int e = blockIdx.x * 8 + (threadIdx.x >> 5);   // exactly N_EDGES waves
    int lane = threadIdx.x & 31;
    int s = ei[e];
    int d = ei[N_EDGES + e];
    int c0 = lane * 4;
    const float4 v = *(const float4*)(x + (long)s * DPAD + c0);
    float* dst = agg + (long)d * DPAD + c0;
    if (c0 + 3 < D) {
        atomicAdd(dst + 0, v.x); atomicAdd(dst + 1, v.y);
        atomicAdd(dst + 2, v.z); atomicAdd(dst + 3, v.w);
    } else {
        if (c0 + 0 < D) atomicAdd(dst + 0, v.x);
        if (c0 + 1 < D) atomicAdd(dst + 1, v.y);
        if (c0 + 2 < D) atomicAdd(dst + 2, v.z);
    }
}

// ----------------------------------------------------------- fused SAGE GEMM
// out[m][n] = sum_k agg[m][k]*dinv[m]*Wl[n][k] + bl[n] + sum_k x[m][k]*Wr[n][k]
// 8 waves/block, one 16x128 output tile per wave, v_wmma_f32_16x16x4_f32.
// Writes in place over `agg` (tiles are row-disjoint across waves).
__global__ __launch_bounds__(256)
void sage_gemm(const float* agg, const float* __restrict__ xin,
               const float* __restrict__ wpad, const float* __restrict__ dinv,
               const float* __restrict__ bias, float* out, int layer) {
    __shared__ float Bsh[DPAD * DPAD / 2];        // 32 KB: half of one phase's panel
    const int tid   = threadIdx.x;
    const int wid   = tid >> 5;
    const int lane  = tid & 31;
    const int tile  = blockIdx.x * 8 + wid;       // wave-uniform
    const int khalf = lane >> 4;                  // 0 or 1
    const int mrow  = (tile << 4) + (lane & 15);  // A-fragment row

    v8f acc[8] = {};                              // 16x128 accumulator tile

    for (int phase = 0; phase < 2; ++phase) {
        __syncthreads();                          // previous phase done w/ Bsh
        {   // cooperative LDS fill: panel is pre-swizzled, straight b128 copy
            const float4* wsrc =
                (const float4*)(wpad + ((long)(layer * 2 + phase) << 14));
            float4* wdst = (float4*)Bsh;
#pragma unroll
            for (int i = 0; i < 16; ++i)
                wdst[tid + (i << 8)] = wsrc[tid + (i << 8)];
        }
        __syncthreads();

        if (tile < M_TILES) {
            const float* A    = phase ? xin : agg;
            const float scale = phase ? 1.0f : dinv[mrow];
            const float* arow = A + (long)mrow * DPAD + (khalf << 1);
#pragma unroll 4
            for (int k = 0; k < DPAD; k += 4) {
                const int kk = k + (khalf << 1);          // even
                const float2 av = *(const float2*)(arow + k);
                v2f a; a.x = av.x * scale; a.y = av.y * scale;
                const float* brow = Bsh + ((kk >> 1) << 8) + ((lane & 15) << 1);
#pragma unroll
                for (int nt = 0; nt < 8; ++nt) {
                    const float2 bv = *(const float2*)(brow + (nt << 5));
                    v2f b; b.x = bv.x; b.y = bv.y;
                    acc[nt] = __builtin_amdgcn_wmma_f32_16x16x4_f32(
                        false, a, false, b, (short)0, acc[nt], false, false);
                }
            }
        }
    }

    if (tile < M_TILES) {   // epilogue: C layout = lanes 0-15 M=r, 16-31 M=8+r
        const int mbase = (tile << 4) + ((lane >> 4) << 3);
#pragma unroll
        for (int nt = 0; nt < 8; ++nt) {
            const int n = (nt << 4) + (lane & 15);
            const float bv = (n < D) ? bias[layer * D + n] : 0.0f;
#pragma unroll
            for (int r = 0; r < 8; ++r)
                out[(long)(mbase + r) * DPAD + n] = acc[nt][r] + bv;
        }
    }
}

// ------------------------------------------------------- residue pooling
__global__ void pool_residues(const float* __restrict__ x,
                              const int* __restrict__ r2a,
                              float* __restrict__ out) {
    long t = (long)blockIdx.x * 256 + threadIdx.x;  // exactly N_ATOMS*DPAD
    int col = (int)(t & 127);
    long atom = t >> 7;
    if (col < D)
        atomicAdd(&out[(long)r2a[atom] * D + col], x[atom * DPAD + col]);
}

// ----------------------------------------------------------------- driver
extern "C" void kernel_launch(void* const* d_in, const int* in_sizes, int n_in,
                              void* d_out, int out_size, void* d_ws, size_t ws_size,
                              hipStream_t stream) {
    const float* onehot = (const float*)d_in[1];
    const int*   ei     = (const int*)d_in[2];
    const int*   r2a    = (const int*)d_in[3];
    const float* lnW    = (const float*)d_in[4];
    const float* lnb    = (const float*)d_in[5];
    const float* Wl     = (const float*)d_in[6];
    const float* Wr     = (const float*)d_in[7];
    const float* bl     = (const float*)d_in[8];
    float* out = (float*)d_out;

    char* ws = (char*)d_ws;
    float* bufA = (float*)(ws + 0);                // 100000*128*4 = 51,200,000 B
    float* bufB = (float*)(ws + 51200000);         // same size
    float* wpad = (float*)(ws + 102400000);        // 4*2*16384*4 = 524,288 B
    float* dinv = (float*)(ws + 102924288);        // 400,000 B (deg then 1/deg)

    // One-time prep
    prep_weights<<<512, 256, 0, stream>>>(Wl, Wr, wpad);
    input_proj<<<50000, 256, 0, stream>>>(onehot, lnW, lnb, bufA);
    zero_f4<<<98, 256, 0, stream>>>((float4*)dinv, 25000);
    deg_count<<<3125, 256, 0, stream>>>(ei, dinv);
    deg_inv<<<391, 256, 0, stream>>>(dinv);

    // 4 chained SAGEConv layers, ping-ponging the two 51 MB buffers
    float* x  = bufA;
    float* ag = bufB;
    for (int l = 0; l < N_LAYERS; ++l) {
        zero_f4<<<12500, 256, 0, stream>>>((float4*)ag, 3200000);
        edge_scatter<<<100000, 256, 0, stream>>>(ei, x, ag);
        sage_gemm<<<782, 256, 0, stream>>>(ag, x, wpad, dinv, bl, ag, l);
        float* t = x; x = ag; ag = t;
    }

    // atom -> residue scatter-sum pooling
    zero_f4<<<1405, 256, 0, stream>>>((float4*)out, 359375);
    pool_residues<<<50000, 256, 0, stream>>>(x, r2a, out);
}